// GraphGAT_58282706207011
// MI455X (gfx1250) — compile-verified
//
#include <hip/hip_runtime.h>
#include <hip/hip_bf16.h>
#include <stdint.h>

#define NODES 50000
#define HEADS 8
#define HD    32
#define F     256   // HEADS * HD
#define LDSPAD 72   // LDS panel row stride in floats (72*2 = 16 mod 64 -> no bank conflicts)

typedef float v2f __attribute__((ext_vector_type(2)));
typedef float v8f __attribute__((ext_vector_type(8)));

// ---------------------------------------------------------------------------
// Generic f32 WMMA GEMM:  C[M,Nc] = A[M,Kd] @ W[Kd,Nc] (+bias), row-major.
// 8 waves / block. Each wave computes a 16x64 strip = 4 C tiles of 16x16,
// reusing its A fragment across the 4 tiles. K-loop steps by 4 using
// V_WMMA_F32_16X16X4_F32.
//
// The block's weight panel W[0:Kd, n0:n0+64] is staged ONCE into LDS via the
// gfx1250 async global->LDS path (GLOBAL_LOAD_ASYNC_TO_LDS_B128, ASYNCcnt),
// then all 8 waves read B fragments from LDS for the whole K loop.
//
// A fragment (ISA 32-bit A 16x4 layout): lanes 0-15 -> M rows, VGPR pair holds
//   K=k..k+1 for lanes 0-15 and K=k+2..k+3 for lanes 16-31.
// B fragment (4x16): VGPR0 = row k (lanes 0-15) / row k+2 (lanes 16-31),
//   VGPR1 = row k+1 / k+3.
// C tile: VGPR i holds (M = m0 + i + 8*half, N = n + lane&15).
// ---------------------------------------------------------------------------
__global__ __launch_bounds__(256) void wmma_gemm_f32(
    const float* __restrict__ A, int lda,
    const float* __restrict__ W, int ldw,
    float* __restrict__ C, int ldc, int coff,
    const float* __restrict__ bias,
    int Mr, int Nc, int Kd)
{
  extern __shared__ float ldsW[];                 // [Kd][LDSPAD]

  const int lane = threadIdx.x & 31;
  const int wave = threadIdx.x >> 5;
  const int m0   = blockIdx.x * 128 + wave * 16;
  const int n0   = blockIdx.y * 64;
  const int half = lane >> 4;   // 0 or 1
  const int l16  = lane & 15;

  // ---- async-stage the 64-wide weight panel into LDS (once per block) -----
  {
    const int tid = threadIdx.x;
    const int n4  = Kd * 16;                      // float4 elements in panel
    for (int i4 = tid; i4 < n4; i4 += 256) {
      const int k = i4 >> 4;                      // panel row
      const int j = (i4 & 15) * 4;                // col (float index) in row
      const float* gp = W + (size_t)k * ldw + n0 + j;
      const unsigned loff = (unsigned)(uintptr_t)(ldsW + k * LDSPAD + j);
      asm volatile("global_load_async_to_lds_b128 %0, %1, off"
                   :: "v"(loff), "v"(gp) : "memory");
    }
    asm volatile("s_wait_asynccnt 0x0" ::: "memory");
    __syncthreads();
  }

  int arow = m0 + l16;
  if (arow >= Mr) arow = Mr - 1;                  // clamp loads; stores guarded
  const float* Arow = A + (size_t)arow * lda + half * 2;

  v8f c0 = {}, c1 = {}, c2 = {}, c3 = {};

  for (int k = 0; k < Kd; k += 4) {
    v2f a;
    {
      const float2 t = *reinterpret_cast<const float2*>(Arow + k);
      a.x = t.x; a.y = t.y;
    }
    const float* Lr0 = ldsW + (k + half * 2) * LDSPAD + l16;  // row krow
    const float* Lr1 = Lr0 + LDSPAD;                          // row krow+1
    v2f b0, b1, b2, b3;
    b0.x = Lr0[0];  b0.y = Lr1[0];
    b1.x = Lr0[16]; b1.y = Lr1[16];
    b2.x = Lr0[32]; b2.y = Lr1[32];
    b3.x = Lr0[48]; b3.y = Lr1[48];

    c0 = __builtin_amdgcn_wmma_f32_16x16x4_f32(false, a, false, b0, (short)0, c0, false, false);
    c1 = __builtin_amdgcn_wmma_f32_16x16x4_f32(false, a, false, b1, (short)0, c1, false, false);
    c2 = __builtin_amdgcn_wmma_f32_16x16x4_f32(false, a, false, b2, (short)0, c2, false, false);
    c3 = __builtin_amdgcn_wmma_f32_16x16x4_f32(false, a, false, b3, (short)0, c3, false, false);
  }

  auto store_tile = [&](const v8f& cc, int t) {
    const int logical_col = n0 + t * 16 + l16;    // column within this GEMM
    const int col = coff + logical_col;           // column in C buffer
    const float badd = bias ? bias[logical_col] : 0.0f;
#pragma unroll
    for (int i = 0; i < 8; ++i) {
      const int row = m0 + half * 8 + i;
      if (row < Mr) C[(size_t)row * ldc + col] = cc[i] + badd;
    }
  };
  store_tile(c0, 0); store_tile(c1, 1); store_tile(c2, 2); store_tile(c3, 3);
}

// ---------------------------------------------------------------------------
// Fill helper (grid-stride).
// ---------------------------------------------------------------------------
__global__ void fill_u32(unsigned* __restrict__ p, unsigned v, size_t n) {
  size_t i = (size_t)blockIdx.x * blockDim.x + threadIdx.x;
  const size_t stride = (size_t)gridDim.x * blockDim.x;
  for (; i < n; i += stride) p[i] = v;
}

// Monotonic float<->u32 map for atomic max on signed floats.
__device__ __forceinline__ unsigned f2u_mono(float f) {
  unsigned u = __float_as_uint(f);
  return (u & 0x80000000u) ? ~u : (u | 0x80000000u);
}
__device__ __forceinline__ float u2f_mono(unsigned u) {
  return __uint_as_float((u & 0x80000000u) ? (u ^ 0x80000000u) : ~u);
}

// ---------------------------------------------------------------------------
// Edge scores + segment max. One wave32 per edge; lane = channel d.
// score[e,h] = sum_d attn[h,d] * leaky_relu(fs[src,h,d] + fd[dst,h,d])
// ---------------------------------------------------------------------------
__global__ __launch_bounds__(256) void edge_score_kernel(
    const float* __restrict__ fs, const float* __restrict__ fd,
    const int* __restrict__ src, const int* __restrict__ dst,
    const float* __restrict__ attn,      // [F]
    float* __restrict__ score,           // [E*HEADS]
    unsigned* __restrict__ smax,         // [NODES*HEADS] encoded
    int E)
{
  const int e = blockIdx.x * 8 + (threadIdx.x >> 5);
  if (e >= E) return;
  const int lane = threadIdx.x & 31;
  const int s = src[e], d = dst[e];
  const float* ps = fs + (size_t)s * F + lane;
  const float* pd = fd + (size_t)d * F + lane;
#pragma unroll
  for (int h = 0; h < HEADS; ++h) {
    float v = ps[h * HD] + pd[h * HD];
    v = (v > 0.0f) ? v : 0.2f * v;                 // leaky_relu(0.2)
    float p = v * attn[h * HD + lane];
#pragma unroll
    for (int off = 16; off > 0; off >>= 1)
      p += __shfl_xor(p, off, 32);
    if (lane == h) {
      score[(size_t)e * HEADS + h] = p;
      atomicMax(&smax[(size_t)d * HEADS + h], f2u_mono(p));
    }
  }
}

// ---------------------------------------------------------------------------
// ex = exp(score - smax[dst]); denom[dst] += ex. Thread per (edge, head).
// `score` is overwritten in place with ex.
// ---------------------------------------------------------------------------
__global__ __launch_bounds__(256) void edge_exp_kernel(
    float* __restrict__ score,
    const unsigned* __restrict__ smax,
    const int* __restrict__ dst,
    float* __restrict__ denom,
    int EH)
{
  const int idx = blockIdx.x * blockDim.x + threadIdx.x;
  if (idx >= EH) return;
  const int e = idx >> 3, h = idx & 7;
  const int d = dst[e];
  const float m  = u2f_mono(smax[(size_t)d * HEADS + h]);
  const float ex = __expf(score[idx] - m);
  score[idx] = ex;
  atomicAdd(&denom[(size_t)d * HEADS + h], ex);
}

// ---------------------------------------------------------------------------
// acc[dst,h,d] += (ex/denom[dst,h]) * fs[src,h,d]. One wave32 per edge.
// ---------------------------------------------------------------------------
__global__ __launch_bounds__(256) void edge_aggregate_kernel(
    const float* __restrict__ fs, const float* __restrict__ ex,
    const float* __restrict__ denom,
    const int* __restrict__ src, const int* __restrict__ dst,
    float* __restrict__ acc, int E)
{
  const int e = blockIdx.x * 8 + (threadIdx.x >> 5);
  if (e >= E) return;
  const int lane = threadIdx.x & 31;
  const int s = src[e], d = dst[e];
  const float* ps = fs + (size_t)s * F + lane;
  float* pa = acc + (size_t)d * F + lane;
#pragma unroll
  for (int h = 0; h < HEADS; ++h) {
    const float alpha = ex[(size_t)e * HEADS + h] / denom[(size_t)d * HEADS + h];
    atomicAdd(&pa[h * HD], alpha * ps[h * HD]);
  }
}

// ---------------------------------------------------------------------------
// out[n,d] = mean_h( f(acc[n,h,d] + bias[h,d]) ); f = elu or identity.
// ---------------------------------------------------------------------------
__global__ __launch_bounds__(256) void finalize_kernel(
    const float* __restrict__ acc, const float* __restrict__ bias,
    float* __restrict__ out, int ND, int do_elu)
{
  const int idx = blockIdx.x * blockDim.x + threadIdx.x;
  if (idx >= ND) return;
  const int n = idx >> 5, dch = idx & 31;
  float sum = 0.0f;
#pragma unroll
  for (int h = 0; h < HEADS; ++h) {
    float x = acc[(size_t)n * F + h * HD + dch] + bias[h * HD + dch];
    if (do_elu) x = (x > 0.0f) ? x : (__expf(x) - 1.0f);
    sum += x;
  }
  out[idx] = sum * 0.125f;
}

// ---------------------------------------------------------------------------
// Host driver
// ---------------------------------------------------------------------------
extern "C" void kernel_launch(void* const* d_in, const int* in_sizes, int n_in,
                              void* d_out, int out_size, void* d_ws, size_t ws_size,
                              hipStream_t stream) {
  const float* sm_feats = (const float*)d_in[0];   // [N,256]
  const float* sp_feats = (const float*)d_in[1];   // [N,128]
  const int*   src      = (const int*)d_in[2];     // [E]
  const int*   dst      = (const int*)d_in[3];     // [E]
  const float* T1_w     = (const float*)d_in[4];   // [256,128]
  const float* T1_b     = (const float*)d_in[5];   // [128]
  const float* T2_w     = (const float*)d_in[6];   // [128,128]
  const float* T2_b     = (const float*)d_in[7];   // [128]
  const float* g1_src_w = (const float*)d_in[8];   // [256,256]
  const float* g1_dst_w = (const float*)d_in[9];   // [256,256]
  const float* g1_attn  = (const float*)d_in[10];  // [8,32]
  const float* g1_bias  = (const float*)d_in[11];  // [256]
  const float* g2_src_w = (const float*)d_in[12];  // [32,256]
  const float* g2_dst_w = (const float*)d_in[13];  // [32,256]
  const float* g2_attn  = (const float*)d_in[14];  // [8,32]
  const float* g2_bias  = (const float*)d_in[15];  // [256]

  const int E  = in_sizes[2];
  const int NN = NODES;

  // Workspace layout (floats)
  float* ws      = (float*)d_ws;
  float* in_feat = ws;                              // [NN,256]
  float* fsrc    = in_feat + (size_t)NN * F;        // [NN,256]
  float* fdst    = fsrc    + (size_t)NN * F;        // [NN,256]
  float* acc     = fdst    + (size_t)NN * F;        // [NN,256]
  float* h1      = acc     + (size_t)NN * F;        // [NN,32]
  float* score   = h1      + (size_t)NN * HD;       // [E,8] (ex after exp pass)
  unsigned* smax = (unsigned*)(score + (size_t)E * HEADS);  // [NN,8]
  float* denom   = (float*)(smax + (size_t)NN * HEADS);     // [NN,8]

  auto gemm = [&](const float* A, int lda, const float* W, int ldw,
                  float* C, int ldc, int coff, const float* bias,
                  int Mr, int Nc, int Kd) {
    dim3 g((Mr + 127) / 128, Nc / 64);
    const size_t shmem = (size_t)Kd * LDSPAD * sizeof(float);
    wmma_gemm_f32<<<g, 256, shmem, stream>>>(A, lda, W, ldw, C, ldc, coff, bias, Mr, Nc, Kd);
  };

  const int eb  = (E + 7) / 8;                 // edge kernels: 8 waves/block
  const int ehb = (E * HEADS + 255) / 256;
  const int ndb = (NN * HD + 255) / 256;

  // ---- input projection: in_feat = [sm @ T1 + b1 | sp @ T2 + b2] ----------
  gemm(sm_feats, 256, T1_w, 128, in_feat, F, 0,   T1_b, NN, 128, 256);
  gemm(sp_feats, 128, T2_w, 128, in_feat, F, 128, T2_b, NN, 128, 128);

  // ---- GATv2 layer 1 ------------------------------------------------------
  gemm(in_feat, F, g1_src_w, F, fsrc, F, 0, nullptr, NN, F, F);
  gemm(in_feat, F, g1_dst_w, F, fdst, F, 0, nullptr, NN, F, F);

  fill_u32<<<2048, 256, 0, stream>>>(smax, 0u, (size_t)NN * HEADS * 2);          // smax + denom
  fill_u32<<<2048, 256, 0, stream>>>((unsigned*)acc, 0u, (size_t)NN * F);

  edge_score_kernel<<<eb, 256, 0, stream>>>(fsrc, fdst, src, dst, g1_attn, score, smax, E);
  edge_exp_kernel<<<ehb, 256, 0, stream>>>(score, smax, dst, denom, E * HEADS);
  edge_aggregate_kernel<<<eb, 256, 0, stream>>>(fsrc, score, denom, src, dst, acc, E);
  finalize_kernel<<<ndb, 256, 0, stream>>>(acc, g1_bias, h1, NN * HD, 1);        // elu + head-mean

  // ---- GATv2 layer 2 ------------------------------------------------------
  gemm(h1, HD, g2_src_w, F, fsrc, F, 0, nullptr, NN, F, HD);
  gemm(h1, HD, g2_dst_w, F, fdst, F, 0, nullptr, NN, F, HD);

  fill_u32<<<2048, 256, 0, stream>>>(smax, 0u, (size_t)NN * HEADS * 2);
  fill_u32<<<2048, 256, 0, stream>>>((unsigned*)acc, 0u, (size_t)NN * F);

  edge_score_kernel<<<eb, 256, 0, stream>>>(fsrc, fdst, src, dst, g2_attn, score, smax, E);
  edge_exp_kernel<<<ehb, 256, 0, stream>>>(score, smax, dst, denom, E * HEADS);
  edge_aggregate_kernel<<<eb, 256, 0, stream>>>(fsrc, score, denom, src, dst, acc, E);
  finalize_kernel<<<ndb, 256, 0, stream>>>(acc, g2_bias, (float*)d_out, NN * HD, 0);  // head-mean only
}